// CenterNetDecoder_51951924412588
// MI455X (gfx1250) — compile-verified
//
#include <hip/hip_runtime.h>
#include <stdint.h>

// ---------------- problem constants (match reference) ----------------
#define BATCH    16
#define NCLASS   80
#define HH       256
#define WW       256
#define HWSZ     (HH * WW)           // 65536
#define NPB      (NCLASS * HWSZ)     // 5,242,880 elements per batch
#define TOPK     100
#define SCALE    4.0f

// ---------------- algorithm constants ----------------
#define BINS     4096                // top-12 bits of sortable uint
#define REP      4                   // LDS histogram replication (conflict relief)
#define CHUNKS   64                  // blocks per batch for streaming passes
#define CAND_MAX 2048                // candidate cap per batch (expected ~250)

// Order-preserving float32 -> uint32 map (larger float -> larger uint)
__device__ __forceinline__ uint32_t f2sortable(float f) {
    uint32_t b = __float_as_uint(f);
    return b ^ (uint32_t(int32_t(b) >> 31) | 0x80000000u);
}

// ---------------------------------------------------------------------
// 0) zero the per-batch histograms (workspace is poisoned / replayed)
// ---------------------------------------------------------------------
__global__ void zero_kernel(uint32_t* __restrict__ p, int n) {
    int i = blockIdx.x * 256 + threadIdx.x;
    if (i < n) p[i] = 0u;
}

// ---------------------------------------------------------------------
// 1) Pass A: per-batch 4096-bin histogram of top-12 sortable bits.
//    LDS-privatized x4 (interleaved copies -> adjacent banks), then a
//    sparse global-atomic flush. Memory-bound: reads 335 MB total.
// ---------------------------------------------------------------------
__global__ __launch_bounds__(256) void hist_kernel(const float* __restrict__ x,
                                                   uint32_t* __restrict__ hist) {
    __shared__ uint32_t bins[BINS * REP];          // 64 KB LDS
    for (int i = threadIdx.x; i < BINS * REP; i += 256) bins[i] = 0u;
    __syncthreads();

    const int b     = blockIdx.y;
    const int chunk = blockIdx.x;
    const int per4  = (NPB / 4) / CHUNKS;          // 20480 float4 per block
    const float4* src = (const float4*)(x + (size_t)b * NPB) + (size_t)chunk * per4;
    const uint32_t rep = threadIdx.x & (REP - 1);

#pragma unroll 2
    for (int i = threadIdx.x; i < per4; i += 256) {
        float4 v = src[i];                          // coalesced 16B/lane stream
        atomicAdd(&bins[((f2sortable(v.x) >> 20) << 2) | rep], 1u);
        atomicAdd(&bins[((f2sortable(v.y) >> 20) << 2) | rep], 1u);
        atomicAdd(&bins[((f2sortable(v.z) >> 20) << 2) | rep], 1u);
        atomicAdd(&bins[((f2sortable(v.w) >> 20) << 2) | rep], 1u);
    }
    __syncthreads();

    uint32_t* gh = hist + (size_t)b * BINS;
    for (int i = threadIdx.x; i < BINS; i += 256) {
        uint32_t s = bins[i * 4 + 0] + bins[i * 4 + 1] + bins[i * 4 + 2] + bins[i * 4 + 3];
        if (s) atomicAdd(&gh[i], s);                // ~450 nonzero bins for N(0,1)
    }
}

// ---------------------------------------------------------------------
// 2) Per-batch threshold via cooperative suffix-scan (one block/batch):
//    find largest bin p with suffix_count(p) >= K. Avoids the serial
//    dependent global-load chain (latency killer); all work is in LDS.
// ---------------------------------------------------------------------
__global__ __launch_bounds__(256) void select_kernel(const uint32_t* __restrict__ hist,
                                                     uint32_t* __restrict__ bound,
                                                     int* __restrict__ cnt) {
    __shared__ uint32_t ssum[256 + 1];
    const int b = blockIdx.x;
    const int t = threadIdx.x;
    const uint32_t* h = hist + (size_t)b * BINS;

    // each thread owns 16 consecutive bins
    uint32_t local[16];
    uint32_t localSum = 0;
#pragma unroll
    for (int j = 0; j < 16; ++j) {
        local[j] = h[t * 16 + j];
        localSum += local[j];
    }
    ssum[t] = localSum;
    if (t == 0) ssum[256] = 0;
    __syncthreads();

    // Hillis-Steele inclusive SUFFIX scan over the 256 partials
    for (int off = 1; off < 256; off <<= 1) {
        uint32_t v = (t + off < 256) ? ssum[t + off] : 0u;
        __syncthreads();
        ssum[t] += v;
        __syncthreads();
    }

    // suffix strictly above this thread's range
    uint32_t above = ssum[t + 1];   // ssum[256]==0 sentinel
    if (above < TOPK) {             // crossing can only occur in this thread's bins
        uint32_t running = above;
#pragma unroll
        for (int j = 15; j >= 0; --j) {
            running += local[j];
            if (running >= TOPK) {  // exactly one (t,j) crosses first
                bound[b] = (uint32_t)(t * 16 + j) << 20;
                cnt[b]   = 0;
                break;
            }
        }
    }
}

// ---------------------------------------------------------------------
// 3) Pass B: compact candidates (u >= bound). Data is streamed through
//    the CDNA5 async-tensor path: double-buffered
//    global_load_async_to_lds_b128 (ASYNCcnt) + s_wait_asynccnt, then
//    ds_read_b128 back. Per-wave private LDS regions -> no barriers.
// ---------------------------------------------------------------------
__global__ __launch_bounds__(256) void compact_kernel(const float* __restrict__ x,
                                                      const uint32_t* __restrict__ bound,
                                                      int* __restrict__ cnt,
                                                      uint32_t* __restrict__ cand_u,
                                                      int* __restrict__ cand_idx) {
    __shared__ __align__(16) float stage[2][256 * 4];   // 8 KB, per-thread 16B slots

    const int b     = blockIdx.y;
    const int chunk = blockIdx.x;
    const int per4  = (NPB / 4) / CHUNKS;               // 20480 float4 per block
    const int iters = per4 / 256;                       // 80
    const uint32_t thr = bound[b];

    const char* gbase = (const char*)(x + (size_t)b * NPB) + (size_t)chunk * per4 * 16;
    const int elemBase = chunk * per4 * 4;              // flat index within batch

    uint32_t lds0 = (uint32_t)(uintptr_t)&stage[0][threadIdx.x * 4];
    uint32_t lds1 = (uint32_t)(uintptr_t)&stage[1][threadIdx.x * 4];

    // prologue: async-stage tile 0
    {
        uint64_t ga = (uint64_t)(uintptr_t)(gbase + (size_t)threadIdx.x * 16);
        asm volatile("global_load_async_to_lds_b128 %0, %1, off"
                     :: "v"(lds0), "v"(ga) : "memory");
    }

    for (int t = 0; t < iters; ++t) {
        uint32_t ldsNext = (t & 1) ? lds0 : lds1;
        if (t + 1 < iters) {
            uint64_t ga = (uint64_t)(uintptr_t)(gbase + (size_t)((t + 1) * 256 + threadIdx.x) * 16);
            asm volatile("global_load_async_to_lds_b128 %0, %1, off"
                         :: "v"(ldsNext), "v"(ga) : "memory");
            asm volatile("s_wait_asynccnt 0x1" ::: "memory");   // current tile landed
        } else {
            asm volatile("s_wait_asynccnt 0x0" ::: "memory");
        }

        const float4* sp = (const float4*)((t & 1) ? &stage[1][threadIdx.x * 4]
                                                   : &stage[0][threadIdx.x * 4]);
        float4 v = *sp;                                  // ds_read_b128
        const int bi = elemBase + (t * 256 + threadIdx.x) * 4;
        float vals[4] = {v.x, v.y, v.z, v.w};
#pragma unroll
        for (int j = 0; j < 4; ++j) {
            uint32_t u = f2sortable(vals[j]);
            if (u >= thr) {                              // rare (~250 per 5.24M)
                int p = atomicAdd(&cnt[b], 1);
                if (p < CAND_MAX) {
                    cand_u[b * CAND_MAX + p]   = u;
                    cand_idx[b * CAND_MAX + p] = bi + j;
                }
            }
        }
    }
}

// ---------------------------------------------------------------------
// 4) Final: per batch, bitonic-sort candidates by (value desc, idx asc)
//    exactly as jax.lax.top_k tie-breaks, decode, gather reg/wh, emit.
// ---------------------------------------------------------------------
__global__ __launch_bounds__(1024) void final_kernel(const float* __restrict__ wh,
                                                     const float* __restrict__ reg,
                                                     const uint32_t* __restrict__ cand_u,
                                                     const int* __restrict__ cand_idx,
                                                     const int* __restrict__ cnt,
                                                     float* __restrict__ out) {
    __shared__ uint64_t key[CAND_MAX];                   // 16 KB
    const int b = blockIdx.x;
    const int n = min(cnt[b], CAND_MAX);

    for (int i = threadIdx.x; i < CAND_MAX; i += 1024) {
        uint64_t k = 0ull;                               // sentinel sorts last
        if (i < n)
            k = ((uint64_t)cand_u[b * CAND_MAX + i] << 32) |
                (uint32_t)~cand_idx[b * CAND_MAX + i];   // tie: smaller idx wins
        key[i] = k;
    }
    __syncthreads();

    for (int kk = 2; kk <= CAND_MAX; kk <<= 1) {
        for (int j = kk >> 1; j > 0; j >>= 1) {
            for (int i = threadIdx.x; i < CAND_MAX; i += 1024) {
                int p = i ^ j;
                if (p > i) {
                    bool desc = ((i & kk) == 0);
                    uint64_t a = key[i], c = key[p];
                    if ((a < c) == desc) { key[i] = c; key[p] = a; }
                }
            }
            __syncthreads();
        }
    }

    if (threadIdx.x < TOPK) {
        const int k = threadIdx.x;
        uint64_t kv = key[k];
        uint32_t u  = (uint32_t)(kv >> 32);
        int idx     = (int)~((uint32_t)kv);
        uint32_t fb = (u & 0x80000000u) ? (u ^ 0x80000000u) : ~u;
        float score = __uint_as_float(fb);

        float cls = (float)idx * (1.0f / 65536.0f);      // true division (float)
        int   s   = idx & (HWSZ - 1);                    // idx % (W*W), H==W
        float ys  = (float)s * (1.0f / 256.0f);          // true division (float)
        int   xs  = s & (WW - 1);

        const float* regb = reg + (size_t)b * 2 * HWSZ;
        const float* whb  = wh  + (size_t)b * 2 * HWSZ;
        float xo = regb[s], yo = regb[HWSZ + s];
        float w  = whb[s],  h  = whb[HWSZ + s];

        float cx = (float)xs + xo;
        float cy = ys + yo;
        float hw2 = w * 0.5f, hh2 = h * 0.5f;

        out[b * TOPK + k]                = cls;          // topk_classes
        out[BATCH * TOPK + b * TOPK + k] = score;        // scores
        float* rb = out + 2 * BATCH * TOPK + ((size_t)b * TOPK + k) * 4;
        rb[0] = (cx - hw2) * SCALE;
        rb[1] = (cy - hh2) * SCALE;
        rb[2] = (cx + hw2) * SCALE;
        rb[3] = (cy + hh2) * SCALE;
    }
}

// ---------------------------------------------------------------------
extern "C" void kernel_launch(void* const* d_in, const int* in_sizes, int n_in,
                              void* d_out, int out_size, void* d_ws, size_t ws_size,
                              hipStream_t stream) {
    const float* x   = (const float*)d_in[0];   // [16,80,256,256]
    const float* wh  = (const float*)d_in[1];   // [16,2,256,256]
    const float* reg = (const float*)d_in[2];   // [16,2,256,256]
    float* out = (float*)d_out;                 // 1600 + 1600 + 6400 floats

    // workspace layout (~512 KB total)
    uint8_t*  ws       = (uint8_t*)d_ws;
    uint32_t* hist     = (uint32_t*)ws;                                    // 16*4096 u32
    uint32_t* bound    = (uint32_t*)(ws + (size_t)BATCH * BINS * 4);       // 16 u32
    int*      cnt      = (int*)     (ws + (size_t)BATCH * BINS * 4 + 64);  // 16 i32
    uint32_t* cand_u   = (uint32_t*)(ws + (size_t)BATCH * BINS * 4 + 128);
    int*      cand_idx = (int*)((uint8_t*)cand_u + (size_t)BATCH * CAND_MAX * 4);

    zero_kernel<<<(BATCH * BINS + 255) / 256, 256, 0, stream>>>(hist, BATCH * BINS);
    hist_kernel<<<dim3(CHUNKS, BATCH), 256, 0, stream>>>(x, hist);
    select_kernel<<<BATCH, 256, 0, stream>>>(hist, bound, cnt);
    compact_kernel<<<dim3(CHUNKS, BATCH), 256, 0, stream>>>(x, bound, cnt, cand_u, cand_idx);
    final_kernel<<<BATCH, 1024, 0, stream>>>(wh, reg, cand_u, cand_idx, cnt, out);
}